// Net3_36335423324474
// MI455X (gfx1250) — compile-verified
//
#include <hip/hip_runtime.h>
#include <hip/hip_bf16.h>

#define N_NODES   50000
#define N_EDGES   800000
#define N_FEAT    128
#define DIM       32
#define N_GRAPHS  500

typedef float v2f __attribute__((ext_vector_type(2)));
typedef float v8f __attribute__((ext_vector_type(8)));

// ---------------------------------------------------------------------------
// Kernel 1: fused node GEMM via V_WMMA_F32_16X16X4_F32.
//   xwrel[N,32] = x @ w_rel
//   base [N,32] = x @ w_root + b_rel
// One wave computes a 16-row tile; 4 accumulators (2 weights x 2 N-halves).
// A-frag (16x4 f32): lane<16 -> M=lane, K={k,k+1}; lane>=16 -> M=lane-16, K={k+2,k+3}
// B-frag (4x16 f32): lane<16 -> K={k,k+1},  N=lane;  lane>=16 -> K={k+2,k+3}, N=lane-16
// D (16x16 f32, 8 VGPRs): vgpr r, lane<16 -> M=r, N=lane; lane>=16 -> M=8+r, N=lane-16
// ---------------------------------------------------------------------------
__global__ void node_gemm(const float* __restrict__ x,
                          const float* __restrict__ w_rel,
                          const float* __restrict__ b_rel,
                          const float* __restrict__ w_root,
                          float* __restrict__ xwrel,
                          float* __restrict__ base) {
  const int lane  = threadIdx.x & 31;
  const int wave  = threadIdx.x >> 5;
  const int tile  = blockIdx.x * (blockDim.x >> 5) + wave;   // 16-row tile id
  const int row0  = tile * 16;
  if (row0 >= N_NODES) return;

  const int m     = lane & 15;          // M (A) or N (B/D) within half
  const int khalf = (lane >> 4) << 1;   // 0 or 2

  v8f acc_rel0 = {}, acc_rel1 = {}, acc_root0 = {}, acc_root1 = {};

  const float* xrow = x + (size_t)(row0 + m) * N_FEAT + khalf;

  #pragma unroll 4
  for (int k = 0; k < N_FEAT; k += 4) {
    v2f a = *(const v2f*)(xrow + k);

    const float* br = w_rel  + (size_t)(k + khalf) * DIM + m;
    const float* bo = w_root + (size_t)(k + khalf) * DIM + m;
    v2f b_rel0  = { br[0],  br[DIM]      };   // N-half 0
    v2f b_rel1  = { br[16], br[DIM + 16] };   // N-half 1
    v2f b_root0 = { bo[0],  bo[DIM]      };
    v2f b_root1 = { bo[16], bo[DIM + 16] };

    acc_rel0  = __builtin_amdgcn_wmma_f32_16x16x4_f32(false, a, false, b_rel0,
                                                      (short)0, acc_rel0, false, false);
    acc_rel1  = __builtin_amdgcn_wmma_f32_16x16x4_f32(false, a, false, b_rel1,
                                                      (short)0, acc_rel1, false, false);
    acc_root0 = __builtin_amdgcn_wmma_f32_16x16x4_f32(false, a, false, b_root0,
                                                      (short)0, acc_root0, false, false);
    acc_root1 = __builtin_amdgcn_wmma_f32_16x16x4_f32(false, a, false, b_root1,
                                                      (short)0, acc_root1, false, false);
  }

  const int   mbase = row0 + ((lane >> 4) << 3);  // +0 or +8
  const float brl0  = b_rel[m];
  const float brl1  = b_rel[m + 16];
  #pragma unroll
  for (int r = 0; r < 8; ++r) {
    size_t o = (size_t)(mbase + r) * DIM + m;
    xwrel[o]      = acc_rel0[r];
    xwrel[o + 16] = acc_rel1[r];
    base[o]       = acc_root0[r] + brl0;
    base[o + 16]  = acc_root1[r] + brl1;
  }
}

// ---------------------------------------------------------------------------
// Kernel 2: zero the pooled accumulator (must re-init every call).
// ---------------------------------------------------------------------------
__global__ void zero_pooled(float* __restrict__ pooled, int n) {
  int i = blockIdx.x * blockDim.x + threadIdx.x;
  if (i < n) pooled[i] = 0.0f;
}

// ---------------------------------------------------------------------------
// Kernel 3: edge scatter in projected (32-dim) space.
//   base[dst] += edge_weight * xwrel[src]   (wave per edge, lane per feature)
// ---------------------------------------------------------------------------
__global__ void edge_scatter(const float* __restrict__ xwrel,
                             const float* __restrict__ ew,
                             const int*   __restrict__ src,
                             const int*   __restrict__ dst,
                             float* __restrict__ base) {
  int gid = blockIdx.x * blockDim.x + threadIdx.x;
  int e   = gid >> 5;
  int n   = gid & 31;
  if (e >= N_EDGES) return;
  int s = src[e];
  int d = dst[e];
  float v = xwrel[(size_t)s * DIM + n] * ew[e];
  atomicAdd(base + (size_t)d * DIM + n, v);
}

// ---------------------------------------------------------------------------
// Kernel 4: relu + global_add_pool (wave per node, lane per feature).
// ---------------------------------------------------------------------------
__global__ void relu_pool(const float* __restrict__ base,
                          const int*   __restrict__ batch,
                          float* __restrict__ pooled) {
  int gid  = blockIdx.x * blockDim.x + threadIdx.x;
  int node = gid >> 5;
  int n    = gid & 31;
  if (node >= N_NODES) return;
  float v = base[(size_t)node * DIM + n];
  v = v > 0.0f ? v : 0.0f;
  atomicAdd(pooled + (size_t)batch[node] * DIM + n, v);
}

// ---------------------------------------------------------------------------
// Kernel 5: MLP head + 2-class log-softmax (thread per graph, tiny).
// ---------------------------------------------------------------------------
__global__ void head(const float* __restrict__ pooled,
                     const float* __restrict__ w_fc1,
                     const float* __restrict__ b_fc1,
                     const float* __restrict__ w_fc2,
                     const float* __restrict__ b_fc2,
                     float* __restrict__ out) {
  int g = blockIdx.x * blockDim.x + threadIdx.x;
  if (g >= N_GRAPHS) return;

  float p[DIM];
  #pragma unroll
  for (int k = 0; k < DIM; ++k) p[k] = pooled[(size_t)g * DIM + k];

  float h2[DIM];
  #pragma unroll 4
  for (int j = 0; j < DIM; ++j) {
    float s = b_fc1[j];
    #pragma unroll
    for (int k = 0; k < DIM; ++k) s += p[k] * w_fc1[k * DIM + j];
    h2[j] = s > 0.0f ? s : 0.0f;
  }

  float l0 = b_fc2[0], l1 = b_fc2[1];
  #pragma unroll
  for (int k = 0; k < DIM; ++k) {
    l0 += h2[k] * w_fc2[k * 2 + 0];
    l1 += h2[k] * w_fc2[k * 2 + 1];
  }
  float mx  = fmaxf(l0, l1);
  float lse = mx + __logf(__expf(l0 - mx) + __expf(l1 - mx));
  out[g * 2 + 0] = l0 - lse;
  out[g * 2 + 1] = l1 - lse;
}

// ---------------------------------------------------------------------------
extern "C" void kernel_launch(void* const* d_in, const int* in_sizes, int n_in,
                              void* d_out, int out_size, void* d_ws, size_t ws_size,
                              hipStream_t stream) {
  const float* x     = (const float*)d_in[0];   // [N_NODES, 128]
  const float* ew    = (const float*)d_in[1];   // [N_EDGES]
  const float* w_rel = (const float*)d_in[2];   // [128, 32]
  const float* b_rel = (const float*)d_in[3];   // [32]
  const float* w_rt  = (const float*)d_in[4];   // [128, 32]
  const float* w_fc1 = (const float*)d_in[5];   // [32, 32]
  const float* b_fc1 = (const float*)d_in[6];   // [32]
  const float* w_fc2 = (const float*)d_in[7];   // [32, 2]
  const float* b_fc2 = (const float*)d_in[8];   // [2]
  const int*   eidx  = (const int*)d_in[9];     // [2, N_EDGES]
  const int*   batch = (const int*)d_in[10];    // [N_NODES]
  float*       out   = (float*)d_out;           // [N_GRAPHS, 2]

  const int* src = eidx;
  const int* dst = eidx + N_EDGES;

  // Workspace partition (fp32): xwrel | base | pooled
  char* ws = (char*)d_ws;
  float* xwrel  = (float*)(ws);
  float* base   = (float*)(ws + (size_t)N_NODES * DIM * sizeof(float));
  float* pooled = (float*)(ws + (size_t)2 * N_NODES * DIM * sizeof(float));

  // 1) Fused WMMA GEMM: xwrel = x@w_rel ; base = x@w_root + b_rel
  {
    int tiles = (N_NODES + 15) / 16;              // 3125
    int wavesPerBlock = 8;                        // 256 threads
    int blocks = (tiles + wavesPerBlock - 1) / wavesPerBlock;
    node_gemm<<<blocks, 256, 0, stream>>>(x, w_rel, b_rel, w_rt, xwrel, base);
  }

  // 2) Zero pooled accumulator
  {
    int n = N_GRAPHS * DIM;
    zero_pooled<<<(n + 255) / 256, 256, 0, stream>>>(pooled, n);
  }

  // 3) Edge scatter-add in 32-dim projected space
  {
    long long total = (long long)N_EDGES * 32;
    int blocks = (int)((total + 255) / 256);
    edge_scatter<<<blocks, 256, 0, stream>>>(xwrel, ew, src, dst, base);
  }

  // 4) ReLU + pool over graphs
  {
    long long total = (long long)N_NODES * 32;
    int blocks = (int)((total + 255) / 256);
    relu_pool<<<blocks, 256, 0, stream>>>(base, batch, pooled);
  }

  // 5) MLP head + log_softmax
  head<<<(N_GRAPHS + 63) / 64, 64, 0, stream>>>(pooled, w_fc1, b_fc1, w_fc2, b_fc2, out);
}